// decoder_33973191311956
// MI455X (gfx1250) — compile-verified
//
#include <hip/hip_runtime.h>
#include <hip/hip_bf16.h>
#include <math.h>

typedef __attribute__((ext_vector_type(2))) float v2f;
typedef __attribute__((ext_vector_type(8))) float v8f;

#define Bsz 64
#define Tt 60
#define Tenc 40
#define CAPL 20
#define Hd 512
#define Ed 512
#define Vv 32000
#define Kih 1024          /* E + H  (x part of K) */
#define NWG_LSTM 32
#define OUT_STEPS 19      /* CAPLEN - 1 */
#define KB 64             /* K-block staged in LDS for the projection GEMM */
#define APAD 68           /* 64 + 4 pad words -> conflict-free ds_load_b64 */

static __device__ __forceinline__ v8f wmma_f32(v2f a, v2f b, v8f c) {
  // D = A(16x4 f32) * B(4x16 f32) + C(16x16 f32)
  return __builtin_amdgcn_wmma_f32_16x16x4_f32(false, a, false, b, (short)0, c,
                                               false, false);
}

static __device__ __forceinline__ float sigmoidf(float x) {
  return 1.0f / (1.0f + __expf(-x));
}

// ---------------------------------------------------------------- init ------
__global__ void init_kernel(unsigned* bar) {
  if (threadIdx.x == 0 && blockIdx.x == 0) *bar = 0u;
}

// --------------------------------------------------------- build xs ---------
// xs[t][b][k], k<512: (t<40 ? 0 : embedding[captions[b][t-40]][k])
//             k>=512: feature[b][t][k-512]
__global__ void build_xs_kernel(const float* __restrict__ feature,
                                const int* __restrict__ captions,
                                const float* __restrict__ embedding,
                                float* __restrict__ xs) {
  unsigned idx = blockIdx.x * blockDim.x + threadIdx.x;
  if (idx >= (unsigned)(Tt * Bsz * Kih)) return;
  unsigned k = idx & (Kih - 1);
  unsigned b = (idx >> 10) & (Bsz - 1);
  unsigned t = idx >> 16;
  float val;
  if (k < Ed) {
    if (t < Tenc) {
      val = 0.0f;
    } else {
      int tok = captions[b * CAPL + (t - Tenc)];
      val = embedding[(size_t)tok * Ed + k];
    }
  } else {
    val = feature[((size_t)b * Tt + t) * Hd + (k - Ed)];
  }
  xs[idx] = val;
}

// ------------------------------------------------------ grid barrier --------
static __device__ __forceinline__ void grid_barrier(unsigned* bar, unsigned target) {
  __syncthreads();
  if (threadIdx.x == 0) {
    __threadfence();
    __hip_atomic_fetch_add(bar, 1u, __ATOMIC_RELEASE, __HIP_MEMORY_SCOPE_AGENT);
    while (__hip_atomic_load(bar, __ATOMIC_ACQUIRE, __HIP_MEMORY_SCOPE_AGENT) < target) {
      __builtin_amdgcn_s_sleep(2);
    }
    __threadfence();
  }
  __syncthreads();
}

// ------------------------------------------------- persistent LSTM ----------
// 32 WGs x 256 threads. WG owns hidden slice j0 = 16*wg. Per step:
//   gates[64][4][16] = xs_t @ W_ih(rows)T + h @ W_hh(rows)T + bias  (WMMA f32)
//   pointwise LSTM update; c kept in LDS, h slice written to h_buf.
__global__ void __launch_bounds__(256)
lstm_kernel(const float* __restrict__ xs,
            const float* __restrict__ W_ih, const float* __restrict__ W_hh,
            const float* __restrict__ b_ih, const float* __restrict__ b_hh,
            float* __restrict__ h_buf, float* __restrict__ dec_h,
            unsigned* bar) {
  __shared__ float g_lds[4][Bsz][16];  // gate, batch, n  (16 KB)
  __shared__ float c_lds[Bsz][16];     // cell state slice (4 KB, persistent)

  const int tid = threadIdx.x;
  const int wg  = blockIdx.x;
  const int j0  = wg * 16;

  // zero c and our h slice
  for (int i = tid; i < Bsz * 16; i += 256) {
    int b = i >> 4, n = i & 15;
    c_lds[b][n] = 0.0f;
    h_buf[b * Hd + j0 + n] = 0.0f;
  }
  grid_barrier(bar, NWG_LSTM);

  const int wave  = tid >> 5;
  const int lane  = tid & 31;
  const int ln    = lane & 15;
  const int khalf = (lane < 16) ? 0 : 2;        // K sub-pair for this half-wave
  const int gate  = wave >> 1;                  // 0..3 = i,f,g,o
  const int m0    = (wave & 1) * 2;             // this wave: m-tiles {m0, m0+1}
  const int r0    = gate * Hd + j0;             // weight row base
  const float bias = b_ih[r0 + ln] + b_hh[r0 + ln];

  const float* wih_row = W_ih + (size_t)(r0 + ln) * Kih + khalf;
  const float* whh_row = W_hh + (size_t)(r0 + ln) * Hd + khalf;
  const int mrowBase = (lane < 16) ? 0 : 8;

  for (int t = 0; t < Tt; ++t) {
    v8f acc0, acc1;
#pragma unroll
    for (int v = 0; v < 8; ++v) { acc0[v] = bias; acc1[v] = bias; }

    // ---- K part 1: x @ W_ih^T  (K = 1024) ----
    const float* x0 = xs + ((size_t)t * Bsz + m0 * 16 + ln) * Kih + khalf;
    const float* x1 = x0 + 16 * Kih;
#pragma unroll 4
    for (int k0 = 0; k0 < Kih; k0 += 4) {
      v2f bv = *(const v2f*)(wih_row + k0);
      v2f a0 = *(const v2f*)(x0 + k0);
      v2f a1 = *(const v2f*)(x1 + k0);
      acc0 = wmma_f32(a0, bv, acc0);
      acc1 = wmma_f32(a1, bv, acc1);
    }
    // ---- K part 2: h @ W_hh^T  (K = 512) ----
    const float* h0 = h_buf + (size_t)(m0 * 16 + ln) * Hd + khalf;
    const float* h1 = h0 + 16 * Hd;
#pragma unroll 4
    for (int k0 = 0; k0 < Hd; k0 += 4) {
      v2f bv = *(const v2f*)(whh_row + k0);
      v2f a0 = *(const v2f*)(h0 + k0);
      v2f a1 = *(const v2f*)(h1 + k0);
      acc0 = wmma_f32(a0, bv, acc0);
      acc1 = wmma_f32(a1, bv, acc1);
    }

    // gates -> LDS (C layout: vgpr v, lanes<16 -> M=v, lanes>=16 -> M=v+8)
#pragma unroll
    for (int v = 0; v < 8; ++v) {
      g_lds[gate][m0 * 16 + mrowBase + v][ln]       = acc0[v];
      g_lds[gate][(m0 + 1) * 16 + mrowBase + v][ln] = acc1[v];
    }
    __syncthreads();

    // pointwise LSTM update for our 16 hidden units x 64 batches
    for (int i = tid; i < Bsz * 16; i += 256) {
      int b = i >> 4, n = i & 15;
      float iv = g_lds[0][b][n];
      float fv = g_lds[1][b][n];
      float gv = g_lds[2][b][n];
      float ov = g_lds[3][b][n];
      float c  = sigmoidf(fv) * c_lds[b][n] + sigmoidf(iv) * tanhf(gv);
      float h  = sigmoidf(ov) * tanhf(c);
      c_lds[b][n] = c;
      h_buf[b * Hd + j0 + n] = h;
      if (t >= Tenc && t < Tenc + OUT_STEPS) {
        // store directly in output-row order: row = b*19 + s
        dec_h[((size_t)b * OUT_STEPS + (t - Tenc)) * Hd + j0 + n] = h;
      }
    }
    grid_barrier(bar, (unsigned)(t + 2) * NWG_LSTM);
  }
}

// ------------------------------------------------- output projection --------
// out[1216][32000] = dec_h[1216][512] @ W_lin[32000][512]^T + b_lin
// grid (250, 19): WG -> 64 rows x 128 cols; wave w -> 16-col strip, 4 M-tiles.
// A (dec_h) block is staged through LDS once per WG (coalesced b64 loads),
// then read conflict-free with ds_load_b64; B (W_lin) rows stream from L2
// with global_prefetch ahead of use.
__global__ void __launch_bounds__(256)
proj_kernel(const float* __restrict__ dec_h, const float* __restrict__ W_lin,
            const float* __restrict__ b_lin, float* __restrict__ out) {
  __shared__ float a_lds[64][APAD];   // 64 rows x 64 K-block, padded (17.4 KB)

  const int tid   = threadIdx.x;
  const int wave  = tid >> 5;
  const int lane  = tid & 31;
  const int ln    = lane & 15;
  const int khalf = (lane < 16) ? 0 : 2;
  const int col   = blockIdx.x * 128 + wave * 16 + ln;  // 250*128 = 32000
  const int row0  = blockIdx.y * 64;                    // 19*64  = 1216

  const float bias = b_lin[col];
  v8f acc[4];
#pragma unroll
  for (int m = 0; m < 4; ++m)
#pragma unroll
    for (int v = 0; v < 8; ++v) acc[m][v] = bias;

  const float* wrow = W_lin + (size_t)col * Hd + khalf;

  for (int kb = 0; kb < Hd; kb += KB) {
    __syncthreads();
    // stage A block: 64 rows x KB cols, coalesced float2 per thread
    for (int e = tid; e < 64 * (KB / 2); e += 256) {
      int r  = e >> 5;               // 32 float2 per row
      int c2 = (e & 31) << 1;
      *(v2f*)(&a_lds[r][c2]) =
          *(const v2f*)(dec_h + (size_t)(row0 + r) * Hd + kb + c2);
    }
    __syncthreads();
    if (kb + KB < Hd)
      __builtin_prefetch(wrow + kb + KB, 0, 3);  // next B block -> cache

#pragma unroll 4
    for (int kc = 0; kc < KB; kc += 4) {
      v2f bv = *(const v2f*)(wrow + kb + kc);
#pragma unroll
      for (int m = 0; m < 4; ++m) {
        v2f av = *(const v2f*)(&a_lds[16 * m + ln][kc + khalf]);
        acc[m] = wmma_f32(av, bv, acc[m]);
      }
    }
  }

  const int mrowBase = (lane < 16) ? 0 : 8;
#pragma unroll
  for (int m = 0; m < 4; ++m)
#pragma unroll
    for (int v = 0; v < 8; ++v) {
      int row = row0 + m * 16 + mrowBase + v;
      out[(size_t)row * Vv + col] = acc[m][v];
    }
}

// ---------------------------------------------------------------- launch ----
extern "C" void kernel_launch(void* const* d_in, const int* in_sizes, int n_in,
                              void* d_out, int out_size, void* d_ws, size_t ws_size,
                              hipStream_t stream) {
  (void)in_sizes; (void)n_in; (void)out_size; (void)ws_size;
  const float* feature   = (const float*)d_in[0];
  const int*   captions  = (const int*)d_in[1];
  const float* embedding = (const float*)d_in[2];
  const float* W_ih      = (const float*)d_in[3];
  const float* W_hh      = (const float*)d_in[4];
  const float* b_ih      = (const float*)d_in[5];
  const float* b_hh      = (const float*)d_in[6];
  const float* W_lin     = (const float*)d_in[7];
  const float* b_lin     = (const float*)d_in[8];
  float* out = (float*)d_out;

  // workspace layout
  unsigned* bar = (unsigned*)d_ws;                       // 256 B
  float* xs     = (float*)((char*)d_ws + 256);           // 60*64*1024*4 = 15.7 MB
  float* h_buf  = xs + (size_t)Tt * Bsz * Kih;           // 64*512*4
  float* dec_h  = h_buf + (size_t)Bsz * Hd;              // 1216*512*4

  init_kernel<<<1, 1, 0, stream>>>(bar);

  const int nx = Tt * Bsz * Kih;
  build_xs_kernel<<<(nx + 255) / 256, 256, 0, stream>>>(feature, captions,
                                                        embedding, xs);

  lstm_kernel<<<NWG_LSTM, 256, 0, stream>>>(xs, W_ih, W_hh, b_ih, b_hh,
                                            h_buf, dec_h, bar);

  proj_kernel<<<dim3(250, 19), 256, 0, stream>>>(dec_h, W_lin, b_lin, out);
}